// Model_498216206600
// MI455X (gfx1250) — compile-verified
//
#include <hip/hip_runtime.h>
#include <math.h>

#define NFRAG   131072
#define NGENES  2000
#define NCELLS  200
#define NGMB    500
#define EDIM    10
#define ENCD    200
#define NSEG    (NCELLS * NGMB)

typedef float v2f __attribute__((ext_vector_type(2)));
typedef float v8f __attribute__((ext_vector_type(8)));

// ---------------- bucketing: count -> scan -> scatter ----------------

__global__ void zero_counts_kernel(int* counts) {
    int i = blockIdx.x * blockDim.x + threadIdx.x;
    if (i < NGENES) counts[i] = 0;
}

__global__ void count_kernel(const int* __restrict__ gm, int* __restrict__ counts) {
    int f = blockIdx.x * blockDim.x + threadIdx.x;
    if (f < NFRAG) atomicAdd(&counts[gm[f]], 1);
}

__global__ void scan_kernel(const int* __restrict__ counts,
                            int* __restrict__ starts, int* __restrict__ cursor) {
    int acc = 0;
    for (int i = 0; i < NGENES; ++i) {
        starts[i] = acc;
        cursor[i] = acc;
        acc += counts[i];
    }
}

__global__ void scatter_kernel(const int* __restrict__ gm,
                               int* __restrict__ cursor, int* __restrict__ order) {
    int f = blockIdx.x * blockDim.x + threadIdx.x;
    if (f < NFRAG) {
        int pos = atomicAdd(&cursor[gm[f]], 1);
        order[pos] = f;
    }
}

// ---------------- per-gene fragment GEMM via V_WMMA_F32_16X16X4_F32 ----------------
// One block per gene; W1[g] (200x10 fp32) + freq table staged in LDS once.
// Wave tile: A = 16 fragments x 4 enc dims (one sincos per lane per K-step),
// B = W1 4x16 slice from LDS, 50 K-steps -> full-precision fp32 h[16x10].
// s[f] = sum_o sigmoid(h+b1)*w2 (w2 folded before pooling), stored by fragment id.

__launch_bounds__(256)
__global__ void frag_wmma_kernel(const float* __restrict__ coords,
                                 const float* __restrict__ W1,
                                 const float* __restrict__ b1,
                                 const float* __restrict__ w2,
                                 const int*   __restrict__ starts,
                                 const int*   __restrict__ counts,
                                 const int*   __restrict__ order,
                                 float*       __restrict__ sArr) {
    __shared__ float sW1[ENCD * EDIM];   // 8000 B
    __shared__ float sb1[16];
    __shared__ float sw2[16];
    __shared__ float sFreq[64];          // 1000^(-2*(i+1)/50), i = 0..49

    const int g   = blockIdx.x;
    const int cnt = counts[g];
    if (cnt == 0) return;
    const int base0 = starts[g];
    const int tid   = threadIdx.x;

    for (int i = tid; i < ENCD * EDIM; i += 256)
        sW1[i] = W1[(size_t)g * (ENCD * EDIM) + i];
    if (tid < 16) {
        sb1[tid] = (tid < EDIM) ? b1[g * EDIM + tid] : 0.0f;
        sw2[tid] = (tid < EDIM) ? w2[g * EDIM + tid] : 0.0f;
    }
    if (tid < 64)   // freq = 2^(-(i+1) * (2/50) * log2(1000)); pad >=50 harmlessly
        sFreq[tid] = exp2f(-0.3986313714f * (float)((tid < 50 ? tid : 49) + 1));
    __syncthreads();

    const int lane = tid & 31;
    const int wave = tid >> 5;
    const int hi   = lane >> 4;     // 0: lanes 0-15, 1: lanes 16-31
    const int lm   = lane & 15;

    const int bend = base0 + cnt;
    const int nT   = (cnt + 15) >> 4;

    const float w2o   = sw2[lm];
    const float b1o   = sb1[lm];
    const bool  bvalid = (lm < EDIM);

    for (int t = wave; t < nT; t += 8) {
        const int base = base0 + (t << 4);

        // A-row fragment for this lane (clamped for partial tiles)
        int aidx = base + lm;
        aidx = (aidx < bend) ? aidx : (bend - 1);
        const int frag = order[aidx];
        const float2 cc = ((const float2*)coords)[frag];

        v8f acc = {0.0f, 0.0f, 0.0f, 0.0f, 0.0f, 0.0f, 0.0f, 0.0f};

        // Two halves: kb 0..24 uses coord0, kb 25..49 uses coord1 (wave-uniform).
        // Lane's angle index: fi = 2*kk + hi; enc column pair: e0 = 100*h + 4*kk + 2*hi.
        #pragma unroll
        for (int half = 0; half < 2; ++half) {
            const float cv = half ? cc.y : cc.x;
            #pragma unroll 5
            for (int kk = 0; kk < 25; ++kk) {
                const int fi = (kk << 1) + hi;
                const int e0 = half * 100 + (kk << 2) + (hi << 1);
                const float x = cv * sFreq[fi];
                float sn, cs;
                sincosf(x, &sn, &cs);            // e0 even -> sin, e0+1 -> cos
                v2f a; a.x = sn; a.y = cs;

                v2f b;                            // rows K=e0 / K=e0+1, col N=lm
                b.x = bvalid ? sW1[e0 * EDIM + lm]       : 0.0f;
                b.y = bvalid ? sW1[(e0 + 1) * EDIM + lm] : 0.0f;

                acc = __builtin_amdgcn_wmma_f32_16x16x4_f32(
                    false, a, false, b, (short)0, acc, false, false);
            }
        }

        // D layout: VGPR r -> (M=r, N=lm) lanes 0-15, (M=r+8, N=lm) lanes 16-31.
        #pragma unroll
        for (int r = 0; r < 8; ++r) {
            const float hval = acc[r] + b1o;
            float tv = w2o / (1.0f + __expf(-hval));   // sigmoid*w2 (w2o==0 for lm>=10)
            tv += __shfl_xor(tv, 8, 16);
            tv += __shfl_xor(tv, 4, 16);
            tv += __shfl_xor(tv, 2, 16);
            tv += __shfl_xor(tv, 1, 16);
            if (lm == 0) {
                const int M    = r + (hi << 3);
                const int oidx = base + M;
                if (oidx < bend) {
                    const int fo = order[oidx];
                    sArr[fo] = tv;                // unique per fragment -> deterministic
                }
            }
        }
    }
}

// ---------------- deterministic segment sum over sorted cellxgene ix ----------------

__global__ void segsum_kernel(const float* __restrict__ sArr,
                              const int*   __restrict__ lcx,
                              const int*   __restrict__ genes_oi,
                              const float* __restrict__ b2,
                              float*       __restrict__ out) {
    const int seg = blockIdx.x * blockDim.x + threadIdx.x;
    if (seg >= NSEG) return;
    int lo = 0, hiB = NFRAG;          // lower_bound of seg in sorted lcx
    while (lo < hiB) {
        const int mid = (lo + hiB) >> 1;
        if (lcx[mid] < seg) lo = mid + 1; else hiB = mid;
    }
    float acc = 0.0f;
    for (int i = lo; i < NFRAG && lcx[i] == seg; ++i) acc += sArr[i];
    out[seg] = acc + b2[genes_oi[seg % NGMB]];
}

// ---------------- launch ----------------

extern "C" void kernel_launch(void* const* d_in, const int* in_sizes, int n_in,
                              void* d_out, int out_size, void* d_ws, size_t ws_size,
                              hipStream_t stream) {
    (void)in_sizes; (void)n_in; (void)out_size; (void)ws_size;
    const float* coords   = (const float*)d_in[0];
    const int*   gm       = (const int*)  d_in[1];
    const int*   lcx      = (const int*)  d_in[2];
    const int*   genes_oi = (const int*)  d_in[3];
    const float* W1       = (const float*)d_in[4];
    const float* b1       = (const float*)d_in[5];
    const float* w2       = (const float*)d_in[6];
    const float* b2       = (const float*)d_in[7];
    float* out = (float*)d_out;

    int* ws      = (int*)d_ws;
    int* counts  = ws;                 // 2000 (slot padded to 2048)
    int* starts  = ws + 2048;          // 2000
    int* cursor  = ws + 4096;          // 2000
    int* order   = ws + 6144;          // 131072
    float* sArr  = (float*)(ws + 6144 + NFRAG);   // 131072

    zero_counts_kernel<<<(NGENES + 255) / 256, 256, 0, stream>>>(counts);
    count_kernel<<<(NFRAG + 255) / 256, 256, 0, stream>>>(gm, counts);
    scan_kernel<<<1, 1, 0, stream>>>(counts, starts, cursor);
    scatter_kernel<<<(NFRAG + 255) / 256, 256, 0, stream>>>(gm, cursor, order);
    frag_wmma_kernel<<<NGENES, 256, 0, stream>>>(coords, W1, b1, w2,
                                                 starts, counts, order, sArr);
    segsum_kernel<<<(NSEG + 255) / 256, 256, 0, stream>>>(sArr, lcx, genes_oi, b2, out);
}